// TopKRouter_24859270709996
// MI455X (gfx1250) — compile-verified
//
#include <hip/hip_runtime.h>

typedef __attribute__((ext_vector_type(2))) float v2f;
typedef __attribute__((ext_vector_type(4))) float v4f;
typedef __attribute__((ext_vector_type(8))) float v8f;

#define B_ 4
#define T_ 8192
#define C_ 1024
#define E_ 8
#define K_ 2

// 32768 tokens total. Block = 256 threads = 8 wave32 = 128 tokens. Grid = 256.
// Each wave: 16-token x 16-expert tile (experts 8..15 are zero pad), K=1024
// accumulated via V_WMMA_F32_16X16X4_F32 (memory-bound kernel; f32 WMMA is
// far above the ~6us HBM roofline requirement).

__global__ __launch_bounds__(256) void topk_router_wmma(
    const float* __restrict__ x, const float* __restrict__ W,
    float* __restrict__ outP, int* __restrict__ outIdx)
{
    // W rows padded to 1028 floats -> B-fragment ds_load_b64 is bank-conflict
    // free (bank = (4e + k) % 64, distinct per lane). Row 8 is a zero row for
    // the 8 pad-expert lanes (same-address broadcast, conflict-free).
    __shared__ float sW[9 * 1028];          // ~37 KB
    __shared__ float sLog[8][16 * 8];       // per-wave logits transpose scratch

    const int tid = threadIdx.x;

    // Cooperative, coalesced load of W (8x1024 f32) into padded LDS.
    for (int i = tid; i < (E_ * C_) / 4; i += 256) {       // 8 iters/thread
        const int e = (i * 4) / C_;
        const int c = (i * 4) % C_;
        v4f w = *(const v4f*)(W + i * 4);
        *(v4f*)(&sW[e * 1028 + c]) = w;                    // 16B aligned
    }
    for (int i = tid; i < 1028; i += 256) sW[8 * 1028 + i] = 0.0f;
    __syncthreads();

    const int lane = tid & 31;
    const int wave = tid >> 5;
    const int m    = lane & 15;           // A row (=token in tile) / B column
    const int koff = (lane >> 4) << 1;    // K phase: 0 (lanes 0-15), 2 (16-31)

    const long token0 = (long)blockIdx.x * 128 + wave * 16;
    const float* aptr = x + (token0 + m) * C_ + koff;
    const float* bptr = &sW[((m < 8) ? m : 8) * 1028 + koff];

    v8f acc = {};
    for (int kk = 0; kk < C_; kk += 128) {
        __builtin_prefetch(aptr + kk + 512, 0, 1);         // 2KB ahead
#pragma unroll
        for (int k = kk; k < kk + 128; k += 4) {
            v2f a = *(const v2f*)(aptr + k);               // global_load_b64
            v2f b = *(const v2f*)(bptr + k);               // ds_load_b64
            acc = __builtin_amdgcn_wmma_f32_16x16x4_f32(
                false, a, false, b, (short)0, acc, false, false);
        }
    }

    // Transpose 16x16 C tile through LDS (only the 8 real expert columns).
    // C layout: lane holds column N=lane%16; VGPR r -> row r (+8 for lanes>=16).
    float* slog = &sLog[wave][0];
    const int grp = lane >> 4;
    if (m < 8) {
#pragma unroll
        for (int r = 0; r < 8; ++r)
            slog[(r + grp * 8) * 8 + m] = acc[r];
    }
    // Same-wave LDS ops are in-order (ISA 7.3): loads below see stores above.

    if (lane < 16) {
        const int t = lane;
        v4f lo = *(const v4f*)(slog + t * 8);
        v4f hi = *(const v4f*)(slog + t * 8 + 4);
        float l[8] = {lo.x, lo.y, lo.z, lo.w, hi.x, hi.y, hi.z, hi.w};

        float mx = l[0];
#pragma unroll
        for (int e = 1; e < 8; ++e) mx = fmaxf(mx, l[e]);
        float p[8], s = 0.0f;
#pragma unroll
        for (int e = 0; e < 8; ++e) { p[e] = __expf(l[e] - mx); s += p[e]; }
        const float inv = 1.0f / s;
#pragma unroll
        for (int e = 0; e < 8; ++e) p[e] *= inv;

        // top-2, first index wins ties (matches jax.lax.top_k)
        int i0 = 0; float p0 = p[0];
#pragma unroll
        for (int e = 1; e < 8; ++e) if (p[e] > p0) { p0 = p[e]; i0 = e; }
        int i1 = (i0 == 0) ? 1 : 0; float p1 = p[i1];
#pragma unroll
        for (int e = 0; e < 8; ++e)
            if (e != i0 && p[e] > p1) { p1 = p[e]; i1 = e; }

        const long token = token0 + t;
        float o[8];
#pragma unroll
        for (int e = 0; e < 8; ++e)
            o[e] = (e == i0) ? p0 : ((e == i1) ? p1 : 0.0f);

        v4f d0 = {o[0], o[1], o[2], o[3]};
        v4f d1 = {o[4], o[5], o[6], o[7]};
        v4f* dst = (v4f*)(outP + token * 8);
        dst[0] = d0;
        dst[1] = d1;
        outIdx[token * 2 + 0] = i0;
        outIdx[token * 2 + 1] = i1;
    }
}

extern "C" void kernel_launch(void* const* d_in, const int* in_sizes, int n_in,
                              void* d_out, int out_size, void* d_ws, size_t ws_size,
                              hipStream_t stream) {
    const float* x = (const float*)d_in[0];   // (B,T,C) f32
    const float* W = (const float*)d_in[1];   // (E,C)   f32
    float* outP  = (float*)d_out;                         // (B,T,E) f32
    int*   outIdx = (int*)(outP + (long)B_ * T_ * E_);    // (B,T,K) i32

    const int blocks = (B_ * T_) / 128;  // 256
    topk_router_wmma<<<blocks, 256, 0, stream>>>(x, W, outP, outIdx);
}